// SimVQ_62285615726592
// MI455X (gfx1250) — compile-verified
//
#include <hip/hip_runtime.h>
#include <hip/hip_bf16.h>

typedef __attribute__((ext_vector_type(2))) float v2f;
typedef __attribute__((ext_vector_type(8))) float v8f;

#define DIM 512
#define CB  8192
#define BN  16384  /* B*N = 4*4096 */

#define KC 64      /* K chunk staged in LDS */
#define NC 64      /* c columns per chunk (4 wmma tiles) */
#define MB 128     /* x rows per workgroup */
#define LDA 66     /* padded LDS strides (bank-conflict free, 8B aligned) */
#define LDB 66

/* workspace layout in floats */
#define WS_IMPLICIT 0
#define WS_NORMS    (CB * DIM)
#define WS_INDICES  (WS_NORMS + CB)
#define WS_LOSS     (WS_INDICES + BN)

#define WMMA_F32(a, b, c) \
    __builtin_amdgcn_wmma_f32_16x16x4_f32(false, (a), false, (b), (short)0, (c), false, false)

// ---------------------------------------------------------------------------
// implicit[c,d] = sum_k codebook[c,k] * W[d,k]   (8192x512x512, f32 WMMA)
// ---------------------------------------------------------------------------
__global__ __launch_bounds__(256) void gemm1_kernel(const float* __restrict__ cbk,
                                                    const float* __restrict__ W,
                                                    float* __restrict__ impl) {
    const int lane = threadIdx.x & 31;
    const int wv   = threadIdx.x >> 5;
    const int half = lane >> 4, lr = lane & 15;
    const int c0 = (blockIdx.x * 8 + wv) * 16;
    const int d0 = blockIdx.y * 16;

    const float* arow = cbk + (size_t)(c0 + lr) * DIM + 2 * half;
    const float* brow = W   + (size_t)(d0 + lr) * DIM + 2 * half;

    v8f acc = {};
#pragma unroll 8
    for (int k = 0; k < DIM; k += 4) {
        v2f a = *(const v2f*)(arow + k);
        v2f b = *(const v2f*)(brow + k);
        acc = WMMA_F32(a, b, acc);
    }

    const int crow = c0 + (half ? 8 : 0);
    float* out = impl + (size_t)crow * DIM + d0 + lr;
#pragma unroll
    for (int v = 0; v < 8; ++v) out[(size_t)v * DIM] = acc[v];
}

// ---------------------------------------------------------------------------
// norms[c] = ||implicit[c,:]||^2
// ---------------------------------------------------------------------------
__global__ __launch_bounds__(256) void norms_kernel(const float* __restrict__ impl,
                                                    float* __restrict__ norms) {
    const int lane = threadIdx.x & 31;
    const int wv   = threadIdx.x >> 5;
    const int c    = blockIdx.x * 8 + wv;
    const float* row = impl + (size_t)c * DIM;
    float s = 0.f;
#pragma unroll
    for (int i = 0; i < DIM / 32; ++i) {
        float v = row[lane + i * 32];
        s += v * v;
    }
#pragma unroll
    for (int m = 16; m >= 1; m >>= 1) s += __shfl_xor(s, m, 32);
    if (lane == 0) norms[c] = s;
}

// ---------------------------------------------------------------------------
// Fused distance GEMM + argmin:
//   d2[r,c] = norms[c] - 2 * x[r,:].implicit[c,:]  ; idx[r] = argmin_c d2
// 8 waves, 128 rows/block, 64-column chunks, K staged through LDS.
// ---------------------------------------------------------------------------
__global__ __launch_bounds__(256) void argmin_kernel(const float* __restrict__ x,
                                                     const float* __restrict__ impl,
                                                     const float* __restrict__ norms,
                                                     int* __restrict__ idx_out) {
    __shared__ float lA[MB * LDA];
    __shared__ float lB[NC * LDB];

    const int tid  = threadIdx.x;
    const int lane = tid & 31;
    const int wv   = tid >> 5;
    const int half = lane >> 4, lr = lane & 15;
    const int mb   = blockIdx.x * MB;

    float minv[8];
    int   mini[8];
#pragma unroll
    for (int v = 0; v < 8; ++v) { minv[v] = 3.402823466e38f; mini[v] = 0; }

    for (int cb = 0; cb < CB; cb += NC) {
        v8f acc0 = {}, acc1 = {}, acc2 = {}, acc3 = {};

        for (int kb = 0; kb < DIM; kb += KC) {
            __syncthreads();
            // stage A: MB x KC floats (4096 float2, 16 per thread, coalesced)
#pragma unroll
            for (int i = 0; i < 16; ++i) {
                int e  = tid + i * 256;
                int r  = e >> 5, c2 = e & 31;
                v2f val = *(const v2f*)(x + (size_t)(mb + r) * DIM + kb + c2 * 2);
                *(v2f*)(lA + r * LDA + c2 * 2) = val;
            }
            // stage B: NC x KC floats (2048 float2, 8 per thread)
#pragma unroll
            for (int i = 0; i < 8; ++i) {
                int e  = tid + i * 256;
                int r  = e >> 5, c2 = e & 31;
                v2f val = *(const v2f*)(impl + (size_t)(cb + r) * DIM + kb + c2 * 2);
                *(v2f*)(lB + r * LDB + c2 * 2) = val;
            }
            __syncthreads();

            const float* ar = lA + (wv * 16 + lr) * LDA + 2 * half;
            const float* b0 = lB + lr * LDB + 2 * half;
#pragma unroll
            for (int ks = 0; ks < KC; ks += 4) {
                v2f a = *(const v2f*)(ar + ks);
                v2f b;
                b = *(const v2f*)(b0 + ks);            acc0 = WMMA_F32(a, b, acc0);
                b = *(const v2f*)(b0 + 16 * LDB + ks); acc1 = WMMA_F32(a, b, acc1);
                b = *(const v2f*)(b0 + 32 * LDB + ks); acc2 = WMMA_F32(a, b, acc2);
                b = *(const v2f*)(b0 + 48 * LDB + ks); acc3 = WMMA_F32(a, b, acc3);
            }
        }

        // argmin update for this 64-column chunk
        const float n0 = norms[cb + lr];
        const float n1 = norms[cb + 16 + lr];
        const float n2 = norms[cb + 32 + lr];
        const float n3 = norms[cb + 48 + lr];
#pragma unroll
        for (int v = 0; v < 8; ++v) {
            float d;
            d = n0 - 2.f * acc0[v]; if (d < minv[v]) { minv[v] = d; mini[v] = cb + lr; }
            d = n1 - 2.f * acc1[v]; if (d < minv[v]) { minv[v] = d; mini[v] = cb + 16 + lr; }
            d = n2 - 2.f * acc2[v]; if (d < minv[v]) { minv[v] = d; mini[v] = cb + 32 + lr; }
            d = n3 - 2.f * acc3[v]; if (d < minv[v]) { minv[v] = d; mini[v] = cb + 48 + lr; }
        }
    }

    // reduce across the 16 lanes sharing each row (masks 1..8 stay in-half)
#pragma unroll
    for (int v = 0; v < 8; ++v) {
        float mv = minv[v];
        int   mi = mini[v];
#pragma unroll
        for (int m = 1; m <= 8; m <<= 1) {
            float ov = __shfl_xor(mv, m, 32);
            int   oi = __shfl_xor(mi, m, 32);
            if (ov < mv || (ov == mv && oi < mi)) { mv = ov; mi = oi; }
        }
        minv[v] = mv; mini[v] = mi;
    }
    if (lr == 0) {
        const int base = mb + wv * 16 + (half ? 8 : 0);
#pragma unroll
        for (int v = 0; v < 8; ++v) idx_out[base + v] = mini[v];
    }
}

// ---------------------------------------------------------------------------
// Gather quantized rows, write float indices, accumulate SSE for commit loss.
// ---------------------------------------------------------------------------
__global__ __launch_bounds__(256) void gather_kernel(const float* __restrict__ x,
                                                     const float* __restrict__ impl,
                                                     const int* __restrict__ idx,
                                                     float* __restrict__ out_q,
                                                     float* __restrict__ out_idx,
                                                     float* __restrict__ loss_acc) {
    const int lane = threadIdx.x & 31;
    const int wv   = threadIdx.x >> 5;
    const int r    = blockIdx.x * 8 + wv;
    const int code = idx[r];

    const float* src = impl + (size_t)code * DIM;
    const float* xr  = x + (size_t)r * DIM;
    float*       dst = out_q + (size_t)r * DIM;

    float s = 0.f;
#pragma unroll
    for (int j = 0; j < 4; ++j) {
        const int o = lane * 4 + j * 128;
        float4 q  = *(const float4*)(src + o);
        float4 xv = *(const float4*)(xr + o);
        *(float4*)(dst + o) = q;
        float dx = xv.x - q.x, dy = xv.y - q.y, dz = xv.z - q.z, dw = xv.w - q.w;
        s += dx * dx + dy * dy + dz * dz + dw * dw;
    }
#pragma unroll
    for (int m = 16; m >= 1; m >>= 1) s += __shfl_xor(s, m, 32);
    if (lane == 0) {
        atomicAdd(loss_acc, s);
        out_idx[r] = (float)code;
    }
}

__global__ void init_kernel(float* loss_acc) { *loss_acc = 0.f; }

__global__ void fin_kernel(const float* __restrict__ loss_acc,
                           float* __restrict__ out_loss) {
    // 0.25*mean + mean == 1.25 * SSE / (B*N*DIM) numerically
    *out_loss = 1.25f * (*loss_acc) / (float)(BN * DIM);
}

// ---------------------------------------------------------------------------
extern "C" void kernel_launch(void* const* d_in, const int* in_sizes, int n_in,
                              void* d_out, int out_size, void* d_ws, size_t ws_size,
                              hipStream_t stream) {
    (void)in_sizes; (void)n_in; (void)out_size; (void)ws_size;
    const float* x   = (const float*)d_in[0];
    const float* cbk = (const float*)d_in[1];
    const float* W   = (const float*)d_in[2];

    float* ws      = (float*)d_ws;
    float* impl    = ws + WS_IMPLICIT;
    float* norms   = ws + WS_NORMS;
    int*   indices = (int*)(ws + WS_INDICES);
    float* loss    = ws + WS_LOSS;

    float* out      = (float*)d_out;
    float* out_q    = out;                       // [BN, DIM]
    float* out_idx  = out + (size_t)BN * DIM;    // [BN]
    float* out_loss = out_idx + BN;              // [1]

    init_kernel<<<dim3(1), dim3(1), 0, stream>>>(loss);
    gemm1_kernel<<<dim3(CB / 16 / 8, DIM / 16), dim3(256), 0, stream>>>(cbk, W, impl);
    norms_kernel<<<dim3(CB / 8), dim3(256), 0, stream>>>(impl, norms);
    argmin_kernel<<<dim3(BN / MB), dim3(256), 0, stream>>>(x, impl, norms, indices);
    gather_kernel<<<dim3(BN / 8), dim3(256), 0, stream>>>(x, impl, indices, out_q, out_idx, loss);
    fin_kernel<<<dim3(1), dim3(1), 0, stream>>>(loss, out_loss);
}